// TrainableMovingAverage1d_11768210391551
// MI455X (gfx1250) — compile-verified
//
#include <hip/hip_runtime.h>

typedef float v2f __attribute__((ext_vector_type(2)));
typedef float v8f __attribute__((ext_vector_type(8)));

// out_flat[m] = sum_{k<8} x_flat[8m+k] * w[k]
// One wave32 computes 16 contiguous outputs via two chained
// V_WMMA_F32_16X16X4_F32 ops (A = x data, B = w broadcast over N).
__global__ __launch_bounds__(256) void TrainableMovingAverage1d_wmma_kernel(
    const float* __restrict__ x, const float* __restrict__ w,
    float* __restrict__ out, long long n_out) {
  const int lane = threadIdx.x & 31;
  const int wave = threadIdx.x >> 5;
  const int half = lane >> 4;   // 0: lanes 0-15, 1: lanes 16-31
  const int l    = lane & 15;

  // Each wave -> 16 contiguous flat outputs.
  const long long base = ((long long)blockIdx.x * 8 + wave) * 16;
  if (base + 16 > n_out) return;  // wave-uniform; EXEC stays all-1s for WMMA

  const long long m = base + l;   // output row of A handled by this lane

  // A-matrix (16x4 f32): lane(l,half) holds K = 2*half, 2*half+1.
  // Two b64 loads per lane; across the wave they tile a contiguous 512B
  // region (16 outputs * 32B) with full cacheline utilization.
  const float* xp = x + m * 8;
  v2f a1 = *(const v2f*)(xp + 2 * half);      // window taps k = 0..3
  v2f a2 = *(const v2f*)(xp + 4 + 2 * half);  // window taps k = 4..7

  // B-matrix (4x16 f32): B[k][n] = w[k] for all n (same (vgpr,half)->K map).
  const v2f* w2 = (const v2f*)w;
  v2f b1 = w2[half];      // { w[2h],   w[2h+1] }
  v2f b2 = w2[half + 2];  // { w[4+2h], w[5+2h] }

  v8f c = {0.f, 0.f, 0.f, 0.f, 0.f, 0.f, 0.f, 0.f};
  // 8 args: (neg_a, A, neg_b, B, c_mod, C, reuse_a, reuse_b); NEG must be 0 for f32.
  c = __builtin_amdgcn_wmma_f32_16x16x4_f32(false, a1, false, b1, (short)0, c,
                                            false, false);
  c = __builtin_amdgcn_wmma_f32_16x16x4_f32(false, a2, false, b2, (short)0, c,
                                            false, false);

  // D[m][n] is identical across n. Per C/D layout, lane 0 holds
  // out[base+0..7] in c[0..7]; lane 16 holds out[base+8..15].
  if (l == 0) {
    float* o = out + base + 8 * half;
    float4 lo, hi;
    lo.x = c[0]; lo.y = c[1]; lo.z = c[2]; lo.w = c[3];
    hi.x = c[4]; hi.y = c[5]; hi.z = c[6]; hi.w = c[7];
    *(float4*)(o)     = lo;   // 16B aligned: base is a multiple of 16
    *(float4*)(o + 4) = hi;
  }
}

extern "C" void kernel_launch(void* const* d_in, const int* in_sizes, int n_in,
                              void* d_out, int out_size, void* d_ws, size_t ws_size,
                              hipStream_t stream) {
  const float* x = (const float*)d_in[0];   // 4096*8192 fp32
  const float* w = (const float*)d_in[1];   // 8 fp32
  float* out = (float*)d_out;               // 4096*1024 fp32

  const long long n_out = (long long)out_size;          // 4,194,304
  // 16 outputs per wave, 8 waves per 256-thread block -> 128 outputs/block.
  const long long waves  = (n_out + 15) / 16;
  const int blocks = (int)((waves + 7) / 8);            // 32768 for this shape

  TrainableMovingAverage1d_wmma_kernel<<<blocks, 256, 0, stream>>>(x, w, out, n_out);
}